// EPR_36326833390319
// MI455X (gfx1250) — compile-verified
//
#include <hip/hip_runtime.h>
#include <hip/hip_bf16.h>

#define DIM     1024
#define NEXP    4
#define BATCH   8
#define NTOK    4096
#define TOTTOK  (BATCH * NTOK)   // 32768 tokens
#define CAP     1024             // floor(0.25 * 4096)

typedef __attribute__((ext_vector_type(2))) float v2f;
typedef __attribute__((ext_vector_type(8))) float v8f;

// ---------------------------------------------------------------------------
// Kernel 1: logits (WMMA f32 16x16x4) + softmax over E=4 -> probs[tok][4]
// Each wave owns a 16-token tile. A = tokens (16xK) streamed from HBM,
// B = W^T (Kx16) from LDS. Lanes n>=4 read a dedicated all-zero LDS row so
// no per-iteration VALU masking is needed in front of the WMMA.
// 256 WMMAs per tile over K=1024. HBM-bound: 128 MiB @ 23.3 TB/s ~ 5.8 us.
// ---------------------------------------------------------------------------
__global__ void __launch_bounds__(256) router_logits_softmax(
    const float* __restrict__ X,      // [TOTTOK, DIM]
    const float* __restrict__ W,      // [NEXP, DIM]
    const float* __restrict__ bias,   // [NEXP]
    float* __restrict__ probs)        // [TOTTOK, NEXP]
{
    __shared__ float Wl[(NEXP + 1) * DIM];  // 20 KB: 4 expert rows + 1 zero row
    __shared__ float tileL[8 * 64];         // per-wave 16x4 logit tile

    const int tid = threadIdx.x;
    for (int i = tid; i < NEXP * DIM; i += 256) Wl[i] = W[i];
    for (int i = tid; i < DIM; i += 256)        Wl[NEXP * DIM + i] = 0.0f;
    __syncthreads();

    const int wave  = tid >> 5;
    const int lane  = tid & 31;
    const int n     = lane & 15;   // N index (expert col) / M index for A row
    const int khalf = lane >> 4;   // half-wave selects K pair

    const int tile = blockIdx.x * 8 + wave;    // 256 blocks * 8 waves = 2048 tiles
    const int tok0 = tile * 16;

    const float* arow = X + (size_t)(tok0 + n) * DIM;        // A: row M = lane&15
    const float* wrow = Wl + ((n < NEXP) ? n : NEXP) * DIM;  // B: col N (zero row if n>=4)

    v8f c = {};
    for (int k0 = 0; k0 < DIM; k0 += 4) {
        const int kk = k0 + 2 * khalf;                       // K = vgpr + 2*khalf
        float2 av = *(const float2*)(arow + kk);
        float2 wv = *(const float2*)(wrow + kk);
        v2f a = {av.x, av.y};
        v2f b = {wv.x, wv.y};
        c = __builtin_amdgcn_wmma_f32_16x16x4_f32(
                /*neg_a=*/false, a, /*neg_b=*/false, b,
                /*c_mod=*/(short)0, c, /*reuse_a=*/false, /*reuse_b=*/false);
    }

    // D layout: lanes 0-15 -> M=v, lanes 16-31 -> M=v+8; N = lane&15
    if (n < NEXP) {
        #pragma unroll
        for (int v = 0; v < 8; ++v)
            tileL[wave * 64 + (v + 8 * khalf) * 4 + n] = c[v];
    }
    __syncthreads();

    if (lane < 16) {
        const int tok = tok0 + lane;
        float l0 = tileL[wave * 64 + lane * 4 + 0] + bias[0];
        float l1 = tileL[wave * 64 + lane * 4 + 1] + bias[1];
        float l2 = tileL[wave * 64 + lane * 4 + 2] + bias[2];
        float l3 = tileL[wave * 64 + lane * 4 + 3] + bias[3];
        float m  = fmaxf(fmaxf(l0, l1), fmaxf(l2, l3));
        float e0 = __expf(l0 - m), e1 = __expf(l1 - m);
        float e2 = __expf(l2 - m), e3 = __expf(l3 - m);
        float inv = 1.0f / (e0 + e1 + e2 + e3);
        ((float4*)probs)[tok] = make_float4(e0 * inv, e1 * inv, e2 * inv, e3 * inv);
    }
}

// ---------------------------------------------------------------------------
// Kernel 2: per-row exact top-CAP via full descending bitonic sort in LDS.
// Key = (monotone(float) << 32) | (0xFFFFFFFF - idx): ties (incl. -inf from
// masked tokens) resolve to lowest index first, matching jax.lax.top_k.
// One workgroup (1024 threads = 32 waves) per batch row; data is L2-resident.
// ---------------------------------------------------------------------------
__global__ void __launch_bounds__(1024) topk_rows(
    const float* __restrict__ probs,   // [TOTTOK, NEXP]
    const int*   __restrict__ mask,    // [NTOK] shared across rows
    int*         __restrict__ topk,    // [BATCH, CAP]
    int j)
{
    __shared__ unsigned long long keys[NTOK];   // 32 KB
    const int row = blockIdx.x;
    const int tid = threadIdx.x;

    for (int i = tid; i < NTOK; i += 1024) {
        float f = (mask[i] != -1) ? -__builtin_inff()
                                  : probs[((size_t)row * NTOK + i) * NEXP + j];
        unsigned u = __float_as_uint(f);
        u = (u & 0x80000000u) ? ~u : (u | 0x80000000u);   // monotone total order
        keys[i] = ((unsigned long long)u << 32) | (0xFFFFFFFFu - (unsigned)i);
    }
    __syncthreads();

    for (unsigned k = 2; k <= NTOK; k <<= 1) {
        for (unsigned s = k >> 1; s > 0; s >>= 1) {
            for (unsigned i = tid; i < NTOK; i += 1024) {
                unsigned ix = i ^ s;
                if (ix > i) {
                    bool up = ((i & k) != 0);          // overall descending
                    unsigned long long a = keys[i], b = keys[ix];
                    bool sw = up ? (a > b) : (a < b);
                    if (sw) { keys[i] = b; keys[ix] = a; }
                }
            }
            __syncthreads();
        }
    }

    if (tid < CAP)
        topk[row * CAP + tid] =
            (int)(0xFFFFFFFFu - (unsigned)(keys[tid] & 0xFFFFFFFFu));
}

// ---------------------------------------------------------------------------
// Kernel 3: union-scatter mask[idx] = j over all rows' selections.
// All writers store the same value -> races benign. Runs strictly after
// topk_rows of the same round (stream ordering = inter-WG sync).
// ---------------------------------------------------------------------------
__global__ void __launch_bounds__(256) scatter_mask(
    const int* __restrict__ topk, int* __restrict__ mask, int j)
{
    int i = blockIdx.x * blockDim.x + threadIdx.x;
    if (i < BATCH * CAP) mask[topk[i]] = j;
}

__global__ void __launch_bounds__(256) init_mask(int* __restrict__ mask)
{
    int i = blockIdx.x * blockDim.x + threadIdx.x;
    if (i < NTOK) mask[i] = -1;
}

// ---------------------------------------------------------------------------
// Kernel 4: outputs. token_mask (int32 bits) then expert_probs (f32), flat.
// ---------------------------------------------------------------------------
__global__ void __launch_bounds__(256) finalize(
    const float* __restrict__ probs, const int* __restrict__ mask,
    int* __restrict__ tm_out, float* __restrict__ ep_out)
{
    int i = blockIdx.x * blockDim.x + threadIdx.x;
    if (i < TOTTOK) {
        int n = i & (NTOK - 1);
        int t = mask[n];
        t = (t < 0) ? 0 : t;
        tm_out[i] = t;
        ep_out[i] = probs[(size_t)i * NEXP + t];
    }
}

extern "C" void kernel_launch(void* const* d_in, const int* in_sizes, int n_in,
                              void* d_out, int out_size, void* d_ws, size_t ws_size,
                              hipStream_t stream)
{
    (void)in_sizes; (void)n_in; (void)out_size; (void)ws_size;
    const float* X    = (const float*)d_in[0];   // [8, 4096, 1024] f32
    const float* W    = (const float*)d_in[1];   // [4, 1024] f32
    const float* bias = (const float*)d_in[2];   // [4] f32

    // Workspace layout
    char*  ws    = (char*)d_ws;
    float* probs = (float*)(ws);                               // 512 KB
    int*   mask  = (int*)(ws + (size_t)TOTTOK * NEXP * 4);     // 16 KB
    int*   topk  = (int*)(ws + (size_t)TOTTOK * NEXP * 4 + NTOK * 4); // 32 KB

    int*   tm_out = (int*)d_out;               // token_mask as raw int32
    float* ep_out = (float*)d_out + TOTTOK;    // expert_probs

    init_mask<<<(NTOK + 255) / 256, 256, 0, stream>>>(mask);

    // 2048 16-token tiles, 8 waves (one tile each) per 256-thread block
    router_logits_softmax<<<TOTTOK / (16 * 8), 256, 0, stream>>>(X, W, bias, probs);

    for (int j = NEXP - 1; j >= 0; --j) {
        topk_rows<<<BATCH, 1024, 0, stream>>>(probs, mask, topk, j);
        scatter_mask<<<(BATCH * CAP + 255) / 256, 256, 0, stream>>>(topk, mask, j);
    }

    finalize<<<(TOTTOK + 255) / 256, 256, 0, stream>>>(probs, mask, tm_out, ep_out);
}